// Routerv4_85401129714222
// MI455X (gfx1250) — compile-verified
//
#include <hip/hip_runtime.h>
#include <hip/hip_bf16.h>

// Shapes from the reference
#define B_    8
#define H_    12
#define L_    1024
#define D_    768
#define K_    512
#define NROWS (H_ * L_)      // 12288 flattened (h,i) rows per batch
#define NCHUNK 48
#define CHUNK_ROWS (NROWS / NCHUNK)   // 256

typedef __attribute__((ext_vector_type(2))) float v2f;
typedef __attribute__((ext_vector_type(8))) float v8f;

// ---------------------------------------------------------------------------
// K1: column sums of scores via V_WMMA_F32_16X16X4_F32.
//     A (16x4, layout documented: lane->M, VGPR+half->K) carries the data:
//         A[m][k] = scores[r + k][j0 + m]
//     B = all ones (layout-invariant), so D[m,n] = sum_k scores[r+k][j0+m].
//     C accumulates across the 256-row chunk.
// ---------------------------------------------------------------------------
__global__ __launch_bounds__(256) void colsum_wmma_kernel(
    const float* __restrict__ scores, float* __restrict__ partial)
{
    const int jt   = blockIdx.x;   // 0..7  (tile of 128 columns)
    const int b    = blockIdx.y;   // 0..7
    const int ch   = blockIdx.z;   // 0..47
    const int tid  = threadIdx.x;
    const int wave = tid >> 5;
    const int lane = tid & 31;
    const int half = lane >> 4;    // 0: K=0,1  | 1: K=2,3   (A layout)
    const int m    = lane & 15;    // A row index M
    const int j0   = jt * 128 + wave * 16;

    const float* base = scores + (size_t)b * NROWS * L_;
    const int r0 = ch * CHUNK_ROWS;

    v8f c = {0.f, 0.f, 0.f, 0.f, 0.f, 0.f, 0.f, 0.f};
    const v2f bones = {1.0f, 1.0f};   // B = ones in every slot -> layout-proof

    for (int r = r0; r < r0 + CHUNK_ROWS; r += 4) {
        // VGPR0 = A[m][2*half]   = scores[r + 2*half    ][j0+m]
        // VGPR1 = A[m][2*half+1] = scores[r + 2*half + 1][j0+m]
        const float* p = base + (size_t)(r + 2 * half) * L_ + (j0 + m);
        v2f a;
        a.x = p[0];
        a.y = p[L_];
        c = __builtin_amdgcn_wmma_f32_16x16x4_f32(
                /*neg_a=*/false, a, /*neg_b=*/false, bones,
                /*c_mod=*/(short)0, c, /*reuse_a=*/false, /*reuse_b=*/false);
    }

    // D layout: VGPR v, lanes 0-15 -> M=v ; lanes 16-31 -> M=v+8.
    // All n columns of D are identical, so lane 0 / lane 16 hold everything.
    if (m == 0) {
        float* outp = partial + ((size_t)(b * NCHUNK + ch) * L_) + j0 + (half ? 8 : 0);
        outp[0] = c[0]; outp[1] = c[1]; outp[2] = c[2]; outp[3] = c[3];
        outp[4] = c[4]; outp[5] = c[5]; outp[6] = c[6]; outp[7] = c[7];
    }
}

// ---------------------------------------------------------------------------
// K2: deterministic reduce over chunks; apply 1/H mean factor.
// ---------------------------------------------------------------------------
__global__ __launch_bounds__(256) void chunk_reduce_kernel(
    const float* __restrict__ partial, float* __restrict__ importance)
{
    int i = blockIdx.x * blockDim.x + threadIdx.x;   // 0 .. B_*L_-1
    if (i >= B_ * L_) return;
    int b = i >> 10, j = i & (L_ - 1);
    float s = 0.f;
    for (int ch = 0; ch < NCHUNK; ++ch)
        s += partial[((size_t)(b * NCHUNK + ch) << 10) + j];
    importance[i] = s * (1.0f / (float)H_);
}

// ---------------------------------------------------------------------------
// K3: stable top-K selection + ascending index output via prefix sum.
//     rank(j) = #{ j' : v[j'] > v[j]  or  (v[j']==v[j] and j' < j) }
//     matches jax.lax.top_k tie-breaking; selected <=> rank < K.
// ---------------------------------------------------------------------------
__global__ __launch_bounds__(1024) void select_kernel(
    const float* __restrict__ importance, int* __restrict__ sel)
{
    __shared__ float imp[L_];
    __shared__ int   flag[L_];
    const int b = blockIdx.x;
    const int j = threadIdx.x;

    float v = importance[b * L_ + j];
    imp[j] = v;
    __syncthreads();

    int rank = 0;
    for (int t = 0; t < L_; ++t) {
        float u = imp[t];
        rank += (u > v) || (u == v && t < j);
    }
    int f = (rank < K_) ? 1 : 0;
    flag[j] = f;
    __syncthreads();

    if (f) {
        int pos = 0;
        for (int t = 0; t < j; ++t) pos += flag[t];
        sel[b * K_ + pos] = j;     // positions come out sorted ascending by j
    }
}

// ---------------------------------------------------------------------------
// K4: assemble output: [class_token ; gathered tokens] and the mask row.
// ---------------------------------------------------------------------------
__global__ __launch_bounds__(256) void gather_kernel(
    const float* __restrict__ hidden, const float* __restrict__ amask,
    const int* __restrict__ sel, float* __restrict__ out_tok,
    float* __restrict__ out_mask)
{
    const int t = blockIdx.x;          // 0 .. 512
    const int b = blockIdx.y;
    const int src = (t == 0) ? 0 : sel[b * K_ + (t - 1)];

    const float4* srcp = (const float4*)(hidden + ((size_t)b * L_ + src) * D_);
    float4*       dstp = (float4*)(out_tok + ((size_t)b * (K_ + 1) + t) * D_);
    const int tid = threadIdx.x;
    if (tid < D_ / 4) dstp[tid] = srcp[tid];

    if (tid == 0)
        out_mask[b * (K_ + 1) + t] = (t == 0) ? 0.0f : amask[b * L_ + src];
}

// ---------------------------------------------------------------------------
extern "C" void kernel_launch(void* const* d_in, const int* in_sizes, int n_in,
                              void* d_out, int out_size, void* d_ws, size_t ws_size,
                              hipStream_t stream) {
    const float* hidden = (const float*)d_in[0];   // (B, L, D) fp32
    const float* amask  = (const float*)d_in[1];   // (B, 1, 1, L) fp32
    const float* scores = (const float*)d_in[2];   // (B, H, L, L) fp32

    // workspace layout (floats):
    //   [0, B*NCHUNK*L)            partial chunk column sums   (1.5 MB)
    //   [.., +B*L)                 importance                  (32 KB)
    //   then B*K ints              selected indices            (16 KB)
    float* ws         = (float*)d_ws;
    float* partial    = ws;
    float* importance = ws + (size_t)B_ * NCHUNK * L_;
    int*   sel        = (int*)(importance + (size_t)B_ * L_);

    float* out_tok  = (float*)d_out;                          // (B, 513, D)
    float* out_mask = out_tok + (size_t)B_ * (K_ + 1) * D_;   // (B, 1, 1, 513)

    dim3 g1(L_ / 128, B_, NCHUNK);
    colsum_wmma_kernel<<<g1, 256, 0, stream>>>(scores, partial);

    chunk_reduce_kernel<<<(B_ * L_ + 255) / 256, 256, 0, stream>>>(partial, importance);

    select_kernel<<<B_, 1024, 0, stream>>>(importance, sel);

    dim3 g4(K_ + 1, B_);
    gather_kernel<<<g4, 256, 0, stream>>>(hidden, amask, sel, out_tok, out_mask);
}